// ConvolutionalCapsule2D_73899207295181
// MI455X (gfx1250) — compile-verified
//
#include <hip/hip_runtime.h>
#include <math.h>

typedef float v2f __attribute__((ext_vector_type(2)));
typedef float v8f __attribute__((ext_vector_type(8)));

#define EPSV 1e-7f
#define NCJ 32
#define MDIM 16
#define CH 32
#define NSPAT 256
#define NBATCH 8

#define WAVES_PER_BLOCK 4
#define UNITS_PER_WAVE 16
#define UNITS_PER_N (64 * CH)                                   // 64 quads * 32 ch = 2048
#define BLOCKS_PER_N (UNITS_PER_N / (WAVES_PER_BLOCK * UNITS_PER_WAVE)) // 32

#define LDS_J_STRIDE 17
#define LDS_PER_WAVE (4 * NCJ * LDS_J_STRIDE) // 2176 floats per wave

// workspace layout in floats
#define WS_ACC   0                                  // [8][32][33]: Sr, Svr[16], Svv[16]
#define WS_MHAT  (WS_ACC + NBATCH * NCJ * 33)       // [8][32][16]
#define WS_INV2S (WS_MHAT + NBATCH * NCJ * MDIM)    // [8][32][16]  = 1/(2s+eps)
#define WS_COEF  (WS_INV2S + NBATCH * NCJ * MDIM)   // [8][32]
#define WS_LOGA  (WS_COEF + NBATCH * NCJ)           // [8][32]

__global__ void zero_acc_kernel(float* __restrict__ ws) {
  int idx = blockIdx.x * blockDim.x + threadIdx.x;
  if (idx < NBATCH * NCJ * 33) ws[WS_ACC + idx] = 0.0f;
}

__global__ __launch_bounds__(WAVES_PER_BLOCK * 32)
void accum_kernel(const float* __restrict__ pose,
                  const float* __restrict__ act,
                  const float* __restrict__ w,
                  float* __restrict__ ws, int iter) {
  __shared__ float vote_lds[WAVES_PER_BLOCK * LDS_PER_WAVE];
  __shared__ float red_acc[NCJ * 33];

  const int lane = threadIdx.x & 31;
  const int wave = threadIdx.x >> 5;
  const int n = blockIdx.y;
  float* vlds = &vote_lds[wave * LDS_PER_WAVE];

  // zero block-level reduction buffer
  for (int i = threadIdx.x; i < NCJ * 33; i += WAVES_PER_BLOCK * 32) red_acc[i] = 0.0f;
  __syncthreads();

  // per-lane E-step constants (lane == j)
  float mhat[MDIM], inv2s[MDIM];
  float coefj = 0.0f, logaj = 0.0f;
  if (iter > 0) {
#pragma unroll
    for (int m = 0; m < MDIM; ++m) {
      mhat[m]  = ws[WS_MHAT  + (n * NCJ + lane) * MDIM + m];
      inv2s[m] = ws[WS_INV2S + (n * NCJ + lane) * MDIM + m];
    }
    coefj = ws[WS_COEF + n * NCJ + lane];
    logaj = ws[WS_LOGA + n * NCJ + lane];
  }

  float Sr = 0.0f, Svr[MDIM], Svv[MDIM];
#pragma unroll
  for (int m = 0; m < MDIM; ++m) { Svr[m] = 0.0f; Svv[m] = 0.0f; }

  // fixed per-lane WMMA fragment decode
  const int frow  = lane & 15;        // A row (16x4) / B col (4x16)
  const int kbase = (lane >> 4) * 2;  // K pair: lanes 0-15 -> K{0,1}, 16-31 -> K{2,3}
  const int jjA = frow >> 2, aA = frow & 3;
  const int iiB = frow >> 2, dB = frow & 3;

  const int waveGlobal = blockIdx.x * WAVES_PER_BLOCK + wave;

  // unit = c*64 + q  ->  c is FIXED for the whole wave (16 units span 16 quads)
  const int c = waveGlobal >> 2;
  const int qbase = (waveGlobal & 3) * UNITS_PER_WAVE;

  // ---- hoisted A fragments: 8 j-groups of w[c, j] (invariant over the unit loop) ----
  v2f afr[8];
  {
    const float* wc = w + (size_t)c * (NCJ * 16);
#pragma unroll
    for (int g = 0; g < 8; ++g) {
      const float* wrow = wc + (g * 4 + jjA) * 16 + aA * 4 + kbase;
      afr[g].x = wrow[0];
      afr[g].y = wrow[1];
    }
  }

  for (int t = 0; t < UNITS_PER_WAVE; ++t) {
    const int q = qbase + t;

    // ---- B fragment: 4 pose matrices (4 spatial positions) ----
    const int spatial = q * 4 + iiB;
    const float* prow = pose + ((size_t)(n * NSPAT + spatial) * CH + c) * MDIM;
    v2f bfrag;
    bfrag.x = prow[kbase * 4 + dB];
    bfrag.y = prow[(kbase + 1) * 4 + dB];
    __builtin_prefetch(prow + 4 * CH * MDIM, 0, 1); // next quad, same lane role

    // ---- 8 WMMAs: votes for 32 j x 4 spatial x 16 m -> scatter to LDS ----
#pragma unroll
    for (int g = 0; g < 8; ++g) {
      v8f czero = {0.f,0.f,0.f,0.f,0.f,0.f,0.f,0.f};
      v8f d = __builtin_amdgcn_wmma_f32_16x16x4_f32(
          false, afr[g], false, bfrag, (short)0, czero, false, false);
#pragma unroll
      for (int k = 0; k < 8; ++k) {
        int row = k + ((lane >> 4) << 3);
        int col = lane & 15;
        int jd = g * 4 + (row >> 2);
        int md = (row & 3) * 4 + (col & 3);
        int iid = col >> 2;
        vlds[(iid * NCJ + jd) * LDS_J_STRIDE + md] = d[k];
      }
    }
    __syncthreads();

    // ---- E-step + accumulate (lane == j) ----
#pragma unroll 1
    for (int ii = 0; ii < 4; ++ii) {
      float vv[MDIM];
#pragma unroll
      for (int m = 0; m < MDIM; ++m)
        vv[m] = vlds[(ii * NCJ + lane) * LDS_J_STRIDE + m];

      float ai = act[(size_t)(n * NSPAT + q * 4 + ii) * CH + c];
      ai = fmaxf(ai, 0.0f); // ROUTE_MIN clamp

      float r;
      if (iter == 0) {
        r = ai * (1.0f / NCJ); // uniform initial routing
      } else {
        float ex = 0.0f;
#pragma unroll
        for (int m = 0; m < MDIM; ++m) {
          float dm = vv[m] - mhat[m];
          ex += dm * dm * inv2s[m];
        }
        float lp = logaj + coefj - ex; // log a_j + log_p (+EPS const: no effect)
        float mx = lp;
#pragma unroll
        for (int off = 16; off; off >>= 1) mx = fmaxf(mx, __shfl_xor(mx, off, 32));
        float e = __expf(lp - mx);
        float ssum = e;
#pragma unroll
        for (int off = 16; off; off >>= 1) ssum += __shfl_xor(ssum, off, 32);
        r = (e / ssum) * ai;
      }

      Sr += r;
#pragma unroll
      for (int m = 0; m < MDIM; ++m) {
        Svr[m] += r * vv[m];
        Svv[m] += r * vv[m] * vv[m];
      }
    }
    __syncthreads();
  }

  // ---- block-level LDS reduction, then global atomics ----
  atomicAdd(&red_acc[lane * 33 + 0], Sr);
#pragma unroll
  for (int m = 0; m < MDIM; ++m) {
    atomicAdd(&red_acc[lane * 33 + 1 + m], Svr[m]);
    atomicAdd(&red_acc[lane * 33 + 17 + m], Svv[m]);
  }
  __syncthreads();
  for (int i = threadIdx.x; i < NCJ * 33; i += WAVES_PER_BLOCK * 32) {
    atomicAdd(&ws[WS_ACC + n * NCJ * 33 + i], red_acc[i]);
  }
}

__global__ __launch_bounds__(32)
void finish_kernel(const float* __restrict__ beta_v,
                   const float* __restrict__ beta_a,
                   float* __restrict__ ws,
                   float* __restrict__ out, int iter) {
  const int n = blockIdx.x;
  const int j = threadIdx.x; // one wave of 32
  const float* accj = ws + WS_ACC + (n * NCJ + j) * 33;

  float Sr = accj[0];
  float den = Sr + EPSV;
  float den2 = den + EPSV;

  float mj[MDIM], sj[MDIM];
  float costsum = 0.0f, coefsum = 0.0f;
#pragma unroll
  for (int m = 0; m < MDIM; ++m) {
    float Svr = accj[1 + m], Svv = accj[17 + m];
    float mm = Svr / den2;
    float ss = (Svv - 2.0f * mm * Svr + mm * mm * Sr) / den2; // sum r (v-m)^2 / (den+eps)
    mj[m] = mm; sj[m] = ss;
    costsum += beta_v[j * MDIM + m] + logf(sqrtf(ss + EPSV) + EPSV);
    coefsum += logf(6.2831853071795865f * ss + EPSV);
  }
  float cost = costsum * den;

  float msum = cost;
#pragma unroll
  for (int off = 16; off; off >>= 1) msum += __shfl_xor(msum, off, 32);
  float mean = msum * (1.0f / NCJ);
  float dcm = cost - mean;
  float vsum = dcm * dcm;
#pragma unroll
  for (int off = 16; off; off >>= 1) vsum += __shfl_xor(vsum, off, 32);
  float stdv = sqrtf(vsum * (1.0f / NCJ) + EPSV);

  float inv_t = 0.5f + 0.1f * (float)iter;
  float x = inv_t * (beta_a[j] + (mean - cost) / (stdv + EPSV));
  float aj = 1.0f / (1.0f + __expf(-x));

#pragma unroll
  for (int m = 0; m < MDIM; ++m) {
    ws[WS_MHAT  + (n * NCJ + j) * MDIM + m] = mj[m];
    ws[WS_INV2S + (n * NCJ + j) * MDIM + m] = 1.0f / (2.0f * sj[m] + EPSV);
  }
  ws[WS_COEF + n * NCJ + j] = -0.5f * coefsum;
  ws[WS_LOGA + n * NCJ + j] = logf(aj + EPSV);

  if (iter == 2) {
#pragma unroll
    for (int m = 0; m < MDIM; ++m) out[(n * NCJ + j) * MDIM + m] = mj[m];
    out[NBATCH * NCJ * MDIM + n * NCJ + j] = aj;
  }
}

extern "C" void kernel_launch(void* const* d_in, const int* in_sizes, int n_in,
                              void* d_out, int out_size, void* d_ws, size_t ws_size,
                              hipStream_t stream) {
  (void)in_sizes; (void)n_in; (void)out_size; (void)ws_size;
  const float* pose   = (const float*)d_in[0];
  const float* act    = (const float*)d_in[1];
  const float* w      = (const float*)d_in[2];
  const float* beta_v = (const float*)d_in[3];
  const float* beta_a = (const float*)d_in[4];
  float* ws  = (float*)d_ws;
  float* out = (float*)d_out;

  dim3 agrid(BLOCKS_PER_N, NBATCH);
  dim3 ablock(WAVES_PER_BLOCK * 32);

  for (int iter = 0; iter < 3; ++iter) {
    zero_acc_kernel<<<(NBATCH * NCJ * 33 + 255) / 256, 256, 0, stream>>>(ws);
    accum_kernel<<<agrid, ablock, 0, stream>>>(pose, act, w, ws, iter);
    finish_kernel<<<NBATCH, NCJ, 0, stream>>>(beta_v, beta_a, ws, out, iter);
  }
}